// SimplifiedL2_996432412952
// MI455X (gfx1250) — compile-verified
//
#include <hip/hip_runtime.h>
#include <hip/hip_bf16.h>
#include <math.h>

// Problem geometry (from reference).
#define BATCH   4
#define SEQ     8192
#define DIM     1024
#define KSEL    4096            // MEMORY_SIZE == k
#define BS      ((size_t)SEQ * DIM)      // batch stride in elements

typedef __attribute__((ext_vector_type(2))) float v2f;
typedef __attribute__((ext_vector_type(8))) float v8f;

__device__ __forceinline__ float v8_get(v8f v, int i) {
    switch (i) {
        case 0: return v[0];
        case 1: return v[1];
        case 2: return v[2];
        case 3: return v[3];
        case 4: return v[4];
        case 5: return v[5];
        case 6: return v[6];
        default: return v[7];
    }
}

// ---------------------------------------------------------------------------
// Kernel 1: importance[s] = (1/4) * sum_b sqrt( sum_d h[b,s,d]^2 )
//
// One wave (32 lanes) handles a 16-token tile for ONE batch. Sum-of-squares is
// computed as diag( X * X^T ) with V_WMMA_F32_16X16X4_F32 accumulating over
// 256 k-steps. A-matrix 16x4 f32 layout: lane L holds A[L%16][K0+{0,1}] with
// K0 = (L>=16 ? 2 : 0); the symmetric B (=A^T) layout needs identical register
// contents, so the same v2f is passed as both A and B.
//
// C/D 16x16 f32 layout: vgpr r, lane L -> row M = r + 8*(L>=16), col N = L%16.
// Diagonal token t: t<8 -> lane t, vgpr t;  t>=8 -> lane t+16, vgpr t-8.
// Block = 128 threads = 4 waves = 4 batches; batch norms summed via ds_add_f32.
// ---------------------------------------------------------------------------
__global__ __launch_bounds__(128) void importance_kernel(
        const float* __restrict__ h, float* __restrict__ imp) {
    __shared__ float lds_sum[16];

    const int tile  = blockIdx.x;             // 16 tokens per tile
    const int lane  = threadIdx.x & 31;
    const int batch = threadIdx.x >> 5;       // wave id == batch id
    const int tok   = lane & 15;
    const int kOff  = ((lane >> 4) & 1) * 2;  // 0 -> dims {k,k+1}; 1 -> {k+2,k+3}

    if (threadIdx.x < 16) lds_sum[threadIdx.x] = 0.0f;
    __syncthreads();

    const float* row = h + (size_t)batch * BS
                         + (size_t)(tile * 16 + tok) * DIM
                         + kOff;

    v8f acc = {};
    #pragma unroll 8
    for (int k = 0; k < DIM; k += 4) {
        const float2 x = *(const float2*)(row + k);
        v2f a;
        a[0] = x.x;
        a[1] = x.y;
        // D = A * A^T + C : same registers serve as A and B (symmetric layout).
        acc = __builtin_amdgcn_wmma_f32_16x16x4_f32(
                  /*neg_a=*/false, a, /*neg_b=*/false, a,
                  /*c_mod=*/(short)0, acc,
                  /*reuse_a=*/false, /*reuse_b=*/false);
    }

    // Pull out diag(X * X^T) = per-token sum of squares, sqrt, batch-accumulate.
    float ssq   = -1.0f;
    int   token = -1;
    if (lane < 8) {                 // tokens 0..7: lane t, vgpr t
        ssq = v8_get(acc, lane);
        token = lane;
    } else if (lane >= 24) {        // tokens 8..15: lane t+16, vgpr t-8
        ssq = v8_get(acc, lane - 24);
        token = lane - 16;
    }
    if (token >= 0) {
        atomicAdd(&lds_sum[token], sqrtf(ssq));   // ds_add_f32
    }
    __syncthreads();

    if (threadIdx.x < 16) {
        imp[tile * 16 + threadIdx.x] = lds_sum[threadIdx.x] * 0.25f;
    }
}

// ---------------------------------------------------------------------------
// Kernel 2: exact sorted top-4096 of 8192 via full bitonic sort in LDS.
// Key = (monotonic(value) << 32) | (0xFFFFFFFF - index): sorting keys
// descending gives value-descending with lower-index-first tie-break,
// matching jax.lax.top_k. 8192 * 8 B = 64 KB LDS (WGP has 320 KB).
// ---------------------------------------------------------------------------
__global__ __launch_bounds__(1024) void topk_sort_kernel(
        const float* __restrict__ imp, int* __restrict__ idx_out) {
    __shared__ unsigned long long keys[SEQ];

    for (int i = threadIdx.x; i < SEQ; i += 1024) {
        unsigned u = __float_as_uint(imp[i]);
        u = (u & 0x80000000u) ? ~u : (u | 0x80000000u);   // monotonic float bits
        keys[i] = ((unsigned long long)u << 32)
                | (unsigned long long)(0xFFFFFFFFu - (unsigned)i);
    }
    __syncthreads();

    for (int k = 2; k <= SEQ; k <<= 1) {
        for (int j = k >> 1; j > 0; j >>= 1) {
            for (int i = threadIdx.x; i < SEQ; i += 1024) {
                const int ixj = i ^ j;
                if (ixj > i) {
                    const unsigned long long a = keys[i];
                    const unsigned long long b = keys[ixj];
                    const bool desc = ((i & k) == 0);     // final pass: all desc
                    if (desc ? (a < b) : (a > b)) {
                        keys[i]   = b;
                        keys[ixj] = a;
                    }
                }
            }
            __syncthreads();
        }
    }

    for (int i = threadIdx.x; i < KSEL; i += 1024) {
        idx_out[i] = (int)(0xFFFFFFFFu - (unsigned)(keys[i] & 0xFFFFFFFFull));
    }
}

// ---------------------------------------------------------------------------
// Kernel 3: out[j, :] = (1/4) * sum_b h[b, idx[j], :]
// 4096 blocks x 256 threads; float4 per thread per batch (fully coalesced).
// k == MEMORY_SIZE, so the whole output is the gathered mean (memory unused).
// ---------------------------------------------------------------------------
__global__ __launch_bounds__(256) void gather_mean_kernel(
        const float* __restrict__ h, const int* __restrict__ idx,
        float* __restrict__ out) {
    const int j = blockIdx.x;
    const int s = idx[j];
    const int d = threadIdx.x << 2;               // 256 * 4 = 1024 dims

    const size_t ro = (size_t)s * DIM + d;
    const float4 a0 = *(const float4*)(h + 0 * BS + ro);
    const float4 a1 = *(const float4*)(h + 1 * BS + ro);
    const float4 a2 = *(const float4*)(h + 2 * BS + ro);
    const float4 a3 = *(const float4*)(h + 3 * BS + ro);

    float4 r;
    r.x = (a0.x + a1.x + a2.x + a3.x) * 0.25f;
    r.y = (a0.y + a1.y + a2.y + a3.y) * 0.25f;
    r.z = (a0.z + a1.z + a2.z + a3.z) * 0.25f;
    r.w = (a0.w + a1.w + a2.w + a3.w) * 0.25f;

    *(float4*)(out + (size_t)j * DIM + d) = r;
}

extern "C" void kernel_launch(void* const* d_in, const int* in_sizes, int n_in,
                              void* d_out, int out_size, void* d_ws, size_t ws_size,
                              hipStream_t stream) {
    const float* h = (const float*)d_in[0];   // [4, 8192, 1024] f32
    // d_in[1] = memory [4096,1024]: unused — k == MEMORY_SIZE fully overwrites it.
    float* out = (float*)d_out;               // [4096, 1024] f32

    float* imp = (float*)d_ws;                              // 8192 floats
    int*   idx = (int*)((char*)d_ws + SEQ * sizeof(float)); // 4096 ints

    importance_kernel<<<SEQ / 16, 128, 0, stream>>>(h, imp);
    topk_sort_kernel<<<1, 1024, 0, stream>>>(imp, idx);
    gather_mean_kernel<<<KSEL, 256, 0, stream>>>(h, idx, out);
}